// ChamferDistance_13159779795611
// MI455X (gfx1250) — compile-verified
//
#include <hip/hip_runtime.h>
#include <math.h>

typedef __attribute__((ext_vector_type(2))) float v2f;
typedef __attribute__((ext_vector_type(8))) float v8f;

#define NBATCH 8
#define NPTS   8192
#define ROW_TILES_PER_WAVE 2                      // 2 x 16 gt rows per wave
#define ROWS_PER_WG (8 * 16 * ROW_TILES_PER_WAVE) // 8 waves * 32 rows = 256
#define COL_TILES (NPTS / 16)                     // 512 pred tiles

// ---------------------------------------------------------------------------
// Kernel 0: init global column-min array to +inf (bit pattern) each launch.
// ---------------------------------------------------------------------------
__global__ __launch_bounds__(256) void chamfer_init(unsigned int* __restrict__ colmin) {
    int idx = blockIdx.x * 256 + threadIdx.x;
    if (idx < NBATCH * NPTS) colmin[idx] = 0x7F800000u; // +inf
}

// ---------------------------------------------------------------------------
// Kernel 1: tiled pairwise distances via V_WMMA_F32_16X16X4_F32.
//   A tile (16 gts rows):   A[m][0..2] = -2*g,  A[m][3] = |g|^2
//   B tile (16 preds cols): B[0..2][n] = p,     B[3][n]  = 1
//   D[m][n] + |p_n|^2 = |g_m - p_n|^2
// Each wave: 32 rows (2 A tiles) x all 8192 columns.
// Column sweep is unrolled x2 (4 WMMAs / iteration):
//   - two pred-tile b96 loads in flight (prefetched one iteration ahead)
//   - the two fresh per-column distances are min'd together BEFORE touching
//     the loop-carried rmin accumulator, halving the v_max canonicalize ops
//     LLVM inserts on the accumulator phis.
// Tile order is wave-staggered (phase = wave*64) to decorrelate LDS atomics.
//   row mins -> finished in-wave via shfl butterflies, direct store
//   col mins -> both lane-halves fire no-return ds_min_u32 at the same LDS
//               word (f32>=0 compared as uint), merged to global at the end.
// ---------------------------------------------------------------------------
__global__ __launch_bounds__(256) void chamfer_main(const float* __restrict__ preds,
                                                    const float* __restrict__ gts,
                                                    unsigned int* __restrict__ colmin_g,
                                                    float* __restrict__ rowmin_g) {
    __shared__ unsigned int sh_col[NPTS]; // 32 KB of the 320 KB WGP LDS

    const int tid  = threadIdx.x;
    const int wave = tid >> 5;
    const int lane = tid & 31;
    const int half = lane >> 4;  // 0: lanes 0-15, 1: lanes 16-31
    const int n    = lane & 15;  // column-within-tile / row-within-tile index
    const int b    = blockIdx.y;

    for (int i = tid; i < NPTS; i += 256) sh_col[i] = 0x7F800000u;
    __syncthreads();

    const float* gts_b   = gts   + (size_t)b * NPTS * 3;
    const float* preds_b = preds + (size_t)b * NPTS * 3;

    const int wave_row_base = blockIdx.x * ROWS_PER_WG + wave * (16 * ROW_TILES_PER_WAVE);

    // Build A tiles. f32 16x4 A layout: lane(half,n) reg r holds K = 2*half+r, M = n.
    v2f a[ROW_TILES_PER_WAVE];
    v8f rmin[ROW_TILES_PER_WAVE];
#pragma unroll
    for (int t = 0; t < ROW_TILES_PER_WAVE; ++t) {
        const int m = wave_row_base + t * 16 + n;
        const float gx = gts_b[m * 3 + 0];
        const float gy = gts_b[m * 3 + 1];
        const float gz = gts_b[m * 3 + 2];
        const float rx = gx * gx + gy * gy + gz * gz;
        if (half == 0) { a[t].x = -2.0f * gx; a[t].y = -2.0f * gy; }
        else           { a[t].x = -2.0f * gz; a[t].y = rx;         }
#pragma unroll
        for (int v = 0; v < 8; ++v) rmin[t][v] = __builtin_inff();
    }

    // Wave-staggered column sweep, unrolled x2, 1-iteration load pipelining.
    int j0 = (((wave * 64) & (COL_TILES - 1)) * 16) + n; // even tile's column
    int j1 = (j0 + 16) & (NPTS - 1);                     // odd tile's column
    float p0x = preds_b[j0 * 3 + 0], p0y = preds_b[j0 * 3 + 1], p0z = preds_b[j0 * 3 + 2];
    float p1x = preds_b[j1 * 3 + 0], p1y = preds_b[j1 * 3 + 1], p1z = preds_b[j1 * 3 + 2];

    for (int it = 0; it < COL_TILES; it += 2) {
        // Prefetch the next pair of pred tiles (wraps; final redundant loads harmless).
        const int jn0 = (j0 + 32) & (NPTS - 1);
        const int jn1 = (j1 + 32) & (NPTS - 1);
        const float q0x = preds_b[jn0 * 3 + 0], q0y = preds_b[jn0 * 3 + 1], q0z = preds_b[jn0 * 3 + 2];
        const float q1x = preds_b[jn1 * 3 + 0], q1y = preds_b[jn1 * 3 + 1], q1z = preds_b[jn1 * 3 + 2];

        const float ry0 = p0x * p0x + p0y * p0y + p0z * p0z;
        const float ry1 = p1x * p1x + p1y * p1y + p1z * p1z;
        v2f bv0, bv1; // B layout symmetric to A: reg r holds K = 2*half+r, N = n
        if (half == 0) { bv0.x = p0x; bv0.y = p0y;  bv1.x = p1x; bv1.y = p1y;  }
        else           { bv0.x = p0z; bv0.y = 1.0f; bv1.x = p1z; bv1.y = 1.0f; }

        float colm0 = __builtin_inff();
        float colm1 = __builtin_inff();
#pragma unroll
        for (int t = 0; t < ROW_TILES_PER_WAVE; ++t) {
            v8f c = {};
            v8f d0 = __builtin_amdgcn_wmma_f32_16x16x4_f32(
                false, a[t], false, bv0, (short)0, c, false, false);
            v8f d1 = __builtin_amdgcn_wmma_f32_16x16x4_f32(
                false, a[t], false, bv1, (short)0, c, false, false);
#pragma unroll
            for (int v = 0; v < 8; ++v) {
                const float dist0 = d0[v] + ry0;           // |g_m - p_j0|^2
                const float dist1 = d1[v] + ry1;           // |g_m - p_j1|^2
                // fresh-value min first (canonical, no canonicalize insert),
                // then a single accumulator (phi) min.
                const float fresh = fminf(dist0, dist1);
                rmin[t][v] = fminf(rmin[t][v], fresh);
                colm0 = fminf(colm0, dist0);               // min over this half's rows
                colm1 = fminf(colm1, dist1);
            }
        }
        // Both halves hit the same LDS word: atomic composes the 16-row mins.
        // No return value -> no dscnt wait, no branch.
        atomicMin(&sh_col[j0], __float_as_uint(colm0));
        atomicMin(&sh_col[j1], __float_as_uint(colm1));

        j0 = jn0; j1 = jn1;
        p0x = q0x; p0y = q0y; p0z = q0z;
        p1x = q1x; p1y = q1y; p1z = q1z;
    }

    // Finish row mins: butterfly over the 16 lanes of each half.
#pragma unroll
    for (int t = 0; t < ROW_TILES_PER_WAVE; ++t) {
#pragma unroll
        for (int v = 0; v < 8; ++v) {
            float r = rmin[t][v];
            r = fminf(r, __shfl_xor(r, 8, 32));
            r = fminf(r, __shfl_xor(r, 4, 32));
            r = fminf(r, __shfl_xor(r, 2, 32));
            r = fminf(r, __shfl_xor(r, 1, 32));
            if (n == 0) { // C/D layout: lanes 0-15 -> M=v, lanes 16-31 -> M=v+8
                const int row = wave_row_base + t * 16 + half * 8 + v;
                rowmin_g[(size_t)b * NPTS + row] = r;
            }
        }
    }

    // Merge this workgroup's column mins into the global array.
    __syncthreads();
    unsigned int* colb = colmin_g + (size_t)b * NPTS;
    for (int i = tid; i < NPTS; i += 256)
        atomicMin(&colb[i], sh_col[i]);
}

// ---------------------------------------------------------------------------
// Kernel 2: deterministic final reduction (single block, fixed order).
// ---------------------------------------------------------------------------
__global__ __launch_bounds__(256) void chamfer_reduce(const unsigned int* __restrict__ colmin,
                                                      const float* __restrict__ rowmin,
                                                      float* __restrict__ out) {
    __shared__ float sh[256];
    float s = 0.0f;
    for (int i = threadIdx.x; i < NBATCH * NPTS; i += 256) {
        s += rowmin[i];                      // loss_2 contributions
        s += __uint_as_float(colmin[i]);     // loss_1 contributions
    }
    sh[threadIdx.x] = s;
    __syncthreads();
    for (int off = 128; off > 0; off >>= 1) {
        if (threadIdx.x < off) sh[threadIdx.x] += sh[threadIdx.x + off];
        __syncthreads();
    }
    if (threadIdx.x == 0) out[0] = sh[0];
}

// ---------------------------------------------------------------------------
extern "C" void kernel_launch(void* const* d_in, const int* in_sizes, int n_in,
                              void* d_out, int out_size, void* d_ws, size_t ws_size,
                              hipStream_t stream) {
    const float* preds = (const float*)d_in[0]; // [8, 8192, 3]
    const float* gts   = (const float*)d_in[1]; // [8, 8192, 3]

    unsigned int* colmin = (unsigned int*)d_ws;                       // 256 KB
    float*        rowmin = (float*)((char*)d_ws +
                                    (size_t)NBATCH * NPTS * sizeof(unsigned int)); // 256 KB

    chamfer_init<<<(NBATCH * NPTS + 255) / 256, 256, 0, stream>>>(colmin);

    dim3 grid(NPTS / ROWS_PER_WG, NBATCH); // (32, 8) workgroups of 256
    chamfer_main<<<grid, 256, 0, stream>>>(preds, gts, colmin, rowmin);

    chamfer_reduce<<<1, 256, 0, stream>>>(colmin, rowmin, (float*)d_out);
}